// YOLOv5Wrapper_86466281603435
// MI455X (gfx1250) — compile-verified
//
#include <hip/hip_runtime.h>
#include <hip/hip_bf16.h>
#include <stdint.h>

#define NUM_CLASSES 18
#define NUM_ANCH    8400
#define SORT_N      16384
#define TOPK        300
#define TPAD        304          // 19 tiles of 16
#define NTILE       (TPAD / 16)  // 19
#define CONF_TH     0.25f
#define IOU_TH      0.45f
#define NTHREADS    256
#define BLK         64           // per-thread-owned sort block (SORT_N / NTHREADS)

typedef float v2f __attribute__((ext_vector_type(2)));
typedef float v8f __attribute__((ext_vector_type(8)));

// LDS layout (bytes):
//   keys      : 16384 * 8            = 131072
//   boxes     : 304 * 16 (float4)    =   4864
//   areaL     : 304 * 4              =   1216
//   valL      : 304 * 4              =   1216
//   sup       : 304 * 10 * 4         =  12160
//   keepMask  : 16 * 4               =     64
#define SMEM_BYTES (131072 + 4864 + 1216 + 1216 + 12160 + 64)

// Monotone float<->uint packing: larger float -> larger uint.
__device__ __forceinline__ unsigned pack_key_hi(float f) {
    unsigned b = __float_as_uint(f);
    return (b & 0x80000000u) ? ~b : (b | 0x80000000u);
}
__device__ __forceinline__ float unpack_key_hi(unsigned u) {
    unsigned b = (u & 0x80000000u) ? (u & 0x7FFFFFFFu) : ~u;
    return __uint_as_float(b);
}

__device__ __forceinline__ unsigned ballot32(bool p) {
#if __has_builtin(__builtin_amdgcn_ballot_w32)
    return __builtin_amdgcn_ballot_w32(p);
#else
    return (unsigned)__ballot(p);
#endif
}

__device__ __forceinline__ uint64_t pack_score_key(float logit, unsigned idx) {
    float s = 1.0f / (1.0f + __expf(-logit));          // sigmoid
    float m = (s > CONF_TH) ? s : -1.0f;               // reference masking
    return ((uint64_t)pack_key_hi(m) << 32) |
           (uint64_t)(0xFFFFFFFFu - idx);              // tie -> lower idx first
}

__device__ __forceinline__ void cmpswap(uint64_t* keys, unsigned i, unsigned j,
                                        unsigned k) {
    unsigned ixj = i ^ j;
    if (ixj > i) {
        uint64_t a = keys[i], b = keys[ixj];
        bool desc = ((i & k) == 0);
        if (desc ? (a < b) : (a > b)) { keys[i] = b; keys[ixj] = a; }
    }
}

__global__ __launch_bounds__(NTHREADS)
void yolo_nms_kernel(const float* __restrict__ in, float* __restrict__ outp) {
    const int img  = blockIdx.x / NUM_CLASSES;
    const int cls  = blockIdx.x % NUM_CLASSES;
    const int tid  = threadIdx.x;
    const int lane = tid & 31;
    const int wv   = tid >> 5;   // 8 waves per block (wave32)

    extern __shared__ unsigned char smem[];
    uint64_t* keys  = (uint64_t*)smem;
    float4* boxesL  = (float4*)(smem + SORT_N * 8);   // [TPAD] {x1,y1,x2,y2}
    float* areaL    = (float*)(boxesL + TPAD);
    float* valL     = areaL + TPAD;
    unsigned* sup   = (unsigned*)(valL + TPAD);   // [TPAD][10] bitmask, bits j>i with iou>thr
    unsigned* keepMask = sup + TPAD * 10;         // 10 words of keep bits (16 alloc'd)

    const float* logits  = in + ((size_t)img * (4 + NUM_CLASSES) + 4 + cls) * NUM_ANCH;
    const float* boxbase = in + (size_t)img * (4 + NUM_CLASSES) * NUM_ANCH;

    __builtin_prefetch(logits, 0, 1);   // global_prefetch_b8

    // ---- Phase 1: build sort keys (float4 loads); zero suppression state --
    {
        const float4* lg4 = (const float4*)logits;      // 8400 % 4 == 0, 16B aligned
        #pragma unroll 1
        for (int q = tid; q < NUM_ANCH / 4; q += NTHREADS) {
            float4 x = lg4[q];
            unsigned b = 4u * (unsigned)q;
            keys[b + 0] = pack_score_key(x.x, b + 0);
            keys[b + 1] = pack_score_key(x.y, b + 1);
            keys[b + 2] = pack_score_key(x.z, b + 2);
            keys[b + 3] = pack_score_key(x.w, b + 3);
        }
        for (int i = NUM_ANCH + tid; i < SORT_N; i += NTHREADS)
            keys[i] = 0;                                 // pad sorts last
    }
    for (int i = tid; i < TPAD * 10; i += NTHREADS) sup[i] = 0u;
    if (tid < 16) keepMask[tid] = 0u;
    __syncthreads();

    // ---- Phase 2: bitonic sort, descending --------------------------------
    // (a) k = 2..64: every pair stays inside a thread-owned 64-block -> no barriers
    {
        const unsigned base = (unsigned)tid * BLK;
        #pragma unroll 1
        for (unsigned k = 2; k <= BLK; k <<= 1)
            #pragma unroll 1
            for (unsigned j = k >> 1; j > 0; j >>= 1)
                #pragma unroll 4
                for (unsigned ii = 0; ii < BLK; ++ii)
                    cmpswap(keys, base + ii, j, k);
    }
    __syncthreads();
    // (b) k = 128..16384: global passes (j >= 64) with barriers, then local cascade
    #pragma unroll 1
    for (unsigned k = 2 * BLK; k <= SORT_N; k <<= 1) {
        unsigned j = k >> 1;
        #pragma unroll 1
        for (; j >= BLK; j >>= 1) {
            #pragma unroll 4
            for (unsigned i = tid; i < SORT_N; i += NTHREADS)
                cmpswap(keys, i, j, k);
            __syncthreads();
        }
        const unsigned base = (unsigned)tid * BLK;
        #pragma unroll 1
        for (; j > 0; j >>= 1)
            #pragma unroll 4
            for (unsigned ii = 0; ii < BLK; ++ii)
                cmpswap(keys, base + ii, j, k);
        __syncthreads();
    }

    // ---- Phase 3: extract top-300, gather boxes, init keep ----------------
    #pragma unroll 1
    for (int t = tid; t < TPAD; t += NTHREADS) {
        float4 bx = {0.f, 0.f, 0.f, 0.f};
        float v = 0.f, ar = 0.f;
        if (t < TOPK) {
            uint64_t key = keys[t];
            unsigned idx = 0xFFFFFFFFu - (unsigned)(key & 0xFFFFFFFFu); // < 8400 always
            v = unpack_key_hi((unsigned)(key >> 32));
            float cx = boxbase[idx];
            float cy = boxbase[NUM_ANCH + idx];
            float w  = boxbase[2 * NUM_ANCH + idx];
            float h  = boxbase[3 * NUM_ANCH + idx];
            bx.x = cx - 0.5f * w; bx.y = cy - 0.5f * h;
            bx.z = cx + 0.5f * w; bx.w = cy + 0.5f * h;
            ar = fmaxf(bx.z - bx.x, 0.f) * fmaxf(bx.w - bx.y, 0.f);
            if (v > CONF_TH) atomicOr(&keepMask[t >> 5], 1u << (t & 31));
        }
        boxesL[t] = bx;                 // one ds_store_b128
        areaL[t] = ar; valL[t] = v;
    }
    __syncthreads();

    // ---- Phase 4: suppression bitmask; pairwise area-sum via WMMA ---------
    // D[i][j] = area_i + area_j  ==  [area_i, 1, 0, 0](16x4) x [1; area_j; 0; 0](4x16)
    {
        const bool hi = (lane >= 16);
        const int  m  = lane & 15;
        int cnt = 0;
        #pragma unroll 1
        for (int ti = 0; ti < NTILE; ++ti) {
            #pragma unroll 1
            for (int tj = ti; tj < NTILE; ++tj) {   // only tiles that can have j > i
                if ((cnt++ & 7) != wv) continue;    // wave-uniform tile distribution
                // A: 16x4 f32, lanes 0-15 hold K0,K1; lanes 16-31 hold K2,K3 (=0)
                v2f a, b;
                a.x = hi ? 0.0f : areaL[ti * 16 + m];
                a.y = hi ? 0.0f : 1.0f;
                // B: 4x16 f32, VGPR0 = K0/K2 rows, VGPR1 = K1/K3 rows
                b.x = hi ? 0.0f : 1.0f;
                b.y = hi ? 0.0f : areaL[tj * 16 + m];
                v8f c = {0.f, 0.f, 0.f, 0.f, 0.f, 0.f, 0.f, 0.f};
#if __has_builtin(__builtin_amdgcn_wmma_f32_16x16x4_f32)
                v8f d = __builtin_amdgcn_wmma_f32_16x16x4_f32(
                    /*neg_a=*/false, a, /*neg_b=*/false, b,
                    /*c_mod=*/(short)0, c, /*reuse_a=*/false, /*reuse_b=*/false);
#else
                v8f d;
                #pragma unroll
                for (int r = 0; r < 8; ++r)
                    d[r] = areaL[ti * 16 + r + (hi ? 8 : 0)] + areaL[tj * 16 + m];
#endif
                const int j  = tj * 16 + m;               // this lane's column
                const int sh = 16 * (tj & 1);             // 16 col bits -> one word
                // column box invariant across rows: one ds_load_b128 per tile
                const float4 jb = boxesL[j];
                #pragma unroll
                for (int r = 0; r < 8; ++r) {
                    int i = ti * 16 + r + (hi ? 8 : 0);
                    const float4 ib = boxesL[i];          // one ds_load_b128 per row
                    float ix1 = fmaxf(ib.x, jb.x);
                    float iy1 = fmaxf(ib.y, jb.y);
                    float ix2 = fminf(ib.z, jb.z);
                    float iy2 = fminf(ib.w, jb.w);
                    float inter = fmaxf(ix2 - ix1, 0.f) * fmaxf(iy2 - iy1, 0.f);
                    float uni   = fmaxf(d[r] - inter, 1e-9f);   // area_i+area_j-inter
                    // inter/uni > TH  <=>  inter > TH*uni   (both >= 0, uni > 0)
                    bool pred = (inter > IOU_TH * uni) &&
                                (j > i) && (i < TOPK) && (j < TOPK);
                    unsigned mask = ballot32(pred);       // [15:0]=rows lo, [31:16]=rows hi
                    if (m == 0) {                         // lanes 0 and 16 emit the ORs
                        unsigned half = hi ? (mask >> 16) : (mask & 0xFFFFu);
                        atomicOr(&sup[i * 10 + (tj >> 1)], half << sh);
                    }
                }
            }
        }
    }
    __syncthreads();

    // ---- Phase 5: greedy NMS on a single wave32 (keep bits in registers) --
    if (wv == 0) {
        unsigned w = (lane < 10) ? keepMask[lane] : 0u;
        #pragma unroll 1
        for (int i = 0; i < TOPK; ++i) {
            unsigned cur = __shfl(w, i >> 5, 32);          // broadcast word holding bit i
            if ((cur >> (i & 31)) & 1u) {                  // wave-uniform branch
                unsigned s = (lane < 10) ? sup[i * 10 + lane] : 0u;
                w &= ~s;                                   // suppress j>i with iou>thr
            }
        }
        if (lane < 10) keepMask[lane] = w;
    }
    __syncthreads();

    // ---- Phase 6: write [x1,y1,x2,y2,score,cls] * keep --------------------
    float* obase = outp + (size_t)(img * NUM_CLASSES + cls) * TOPK * 6;
    #pragma unroll 1
    for (int t = tid; t < TOPK; t += NTHREADS) {
        float mk = ((keepMask[t >> 5] >> (t & 31)) & 1u) ? 1.0f : 0.0f;
        float4 bx = boxesL[t];
        float* o = obase + t * 6;
        o[0] = bx.x * mk;
        o[1] = bx.y * mk;
        o[2] = bx.z * mk;
        o[3] = bx.w * mk;
        o[4] = valL[t] * mk;
        o[5] = (float)cls * mk;
    }
}

extern "C" void kernel_launch(void* const* d_in, const int* in_sizes, int n_in,
                              void* d_out, int out_size, void* d_ws, size_t ws_size,
                              hipStream_t stream) {
    (void)in_sizes; (void)n_in; (void)d_ws; (void)ws_size; (void)out_size;
    const float* in = (const float*)d_in[0];   // (32, 22, 8400) f32
    float* outp     = (float*)d_out;           // (32, 5400, 6) f32
    yolo_nms_kernel<<<dim3(32 * NUM_CLASSES), dim3(NTHREADS), SMEM_BYTES, stream>>>(in, outp);
}